// VectorQuantizer_4509715660932
// MI455X (gfx1250) — compile-verified
//
#include <hip/hip_runtime.h>
#include <hip/hip_bf16.h>

typedef __attribute__((ext_vector_type(2))) float v2f;
typedef __attribute__((ext_vector_type(8))) float v8f;

#define CDIM 512
#define KDIM 8192
#define MROWS 8192
#define ROWS 64    // rows of x per workgroup (four 16-row WMMA tiles per wave)
#define NT 4       // row tiles per wave
#define BLK 512    // threads per workgroup = 16 wave32
#define NWAVE 16
#define LDST 516   // padded LDS row stride (516 % 64 == 4 banks -> conflict-free A reads)

// ---------------------------------------------------------------------------
// Kernel 0: zero histogram + sse accumulator in workspace
// ---------------------------------------------------------------------------
__global__ void vq_zero_kernel(float* __restrict__ hist, float* __restrict__ sse) {
    int i = blockIdx.x * 256 + threadIdx.x;
    if (i < KDIM) hist[i] = 0.0f;
    if (i == 0)  *sse = 0.0f;
}

// ---------------------------------------------------------------------------
// Kernel 1: e2[k] = sum_c emb[c,k]^2   (emb is [C,K] row-major -> coalesced in k)
// ---------------------------------------------------------------------------
__global__ void vq_e2_kernel(const float* __restrict__ emb, float* __restrict__ e2) {
    int k = blockIdx.x * 256 + threadIdx.x;
    float s = 0.0f;
    #pragma unroll 8
    for (int c = 0; c < CDIM; ++c) {
        float v = emb[(size_t)c * KDIM + k];
        s += v * v;
    }
    e2[k] = s;
}

// 64-bit xor-shuffle (wave32) built from two 32-bit shuffles
__device__ __forceinline__ unsigned long long shfl_xor_u64(unsigned long long v, int m) {
    unsigned int lo = (unsigned int)v;
    unsigned int hi = (unsigned int)(v >> 32);
    lo = __shfl_xor(lo, m, 32);
    hi = __shfl_xor(hi, m, 32);
    return ((unsigned long long)hi << 32) | lo;
}

// ---------------------------------------------------------------------------
// Kernel 2: fused distance-GEMM (fp32 WMMA) + argmin + gather + loss + hist
// Grid: MROWS/ROWS workgroups of 512 threads (16 wave32).
// Each wave computes FOUR 16x16 distance tiles (rows 0-63 of the WG's block)
// sharing one B-fetch stream; k-cols = wave's 16-wide slice, sweeping K in
// steps of 16 waves * 16 = 256.
// ---------------------------------------------------------------------------
__global__ __launch_bounds__(BLK) void vq_main_kernel(
    const float* __restrict__ x, const float* __restrict__ emb,
    const float* __restrict__ e2, float* __restrict__ outq,
    float* __restrict__ hist, float* __restrict__ sse) {

    __shared__ float xs[ROWS * LDST];                   // 132096 B
    __shared__ unsigned long long cand[ROWS * NWAVE];   //   8192 B
    __shared__ int rowk[ROWS];
    __shared__ float red[BLK];

    const int tid = threadIdx.x;
    const int m0  = blockIdx.x * ROWS;

    // ---- stage the 64 x-rows into LDS (padded stride) ----
    for (int i = tid; i < ROWS * CDIM; i += BLK) {
        int r = i >> 9;          // /512
        int c = i & (CDIM - 1);
        xs[r * LDST + c] = x[(size_t)(m0 + r) * CDIM + c];
    }
    __syncthreads();

    const int wave  = tid >> 5;
    const int lane  = tid & 31;
    const int laneN = lane & 15;   // N position (B/C/D) and M row (A operand)
    const int hi    = lane >> 4;   // lane half
    const int koff  = hi * 2;      // A/B K-offset within a 4-chunk per ISA f32 layout

    float bestd[NT][8];
    int   bestk[NT][8];
    #pragma unroll
    for (int t = 0; t < NT; ++t)
        #pragma unroll
        for (int r = 0; r < 8; ++r) { bestd[t][r] = 3.4e38f; bestk[t][r] = 0; }

    const float* xrow[NT];
    #pragma unroll
    for (int t = 0; t < NT; ++t)
        xrow[t] = &xs[(16 * t + laneN) * LDST + koff];

    for (int k0 = wave * 16; k0 < KDIM; k0 += NWAVE * 16) {
        const int kk = k0 + laneN;
        const float ek = e2[kk];
        const float* bcol = &emb[(size_t)koff * KDIM + kk];

        v8f acc[NT];
        #pragma unroll
        for (int t = 0; t < NT; ++t)
            acc[t] = (v8f){0.f, 0.f, 0.f, 0.f, 0.f, 0.f, 0.f, 0.f};

        #pragma unroll 2
        for (int c = 0; c < CDIM; c += 4) {
            v2f b;
            b.x = bcol[(size_t)c * KDIM];
            b.y = bcol[(size_t)(c + 1) * KDIM];
            #pragma unroll
            for (int t = 0; t < NT; ++t) {
                v2f a;
                a.x = xrow[t][c];
                a.y = xrow[t][c + 1];
                // D = A(16x4 f32) x B(4x16 f32) + C, full fp32; 4 tiles share B
                acc[t] = __builtin_amdgcn_wmma_f32_16x16x4_f32(
                    false, a, false, b, (short)0, acc[t], false, false);
            }
        }

        // D layout: VGPR r -> row r (lanes 0-15) / row r+8 (lanes 16-31), col = laneN
        #pragma unroll
        for (int t = 0; t < NT; ++t) {
            #pragma unroll
            for (int r = 0; r < 8; ++r) {
                float d = ek - 2.0f * acc[t][r];   // ||e_k||^2 - 2 x.e_k
                if (d < bestd[t][r]) { bestd[t][r] = d; bestk[t][r] = kk; }
            }
        }
    }

    // ---- in-wave argmin: butterfly over the 4 laneN bits on packed u64 ----
    #pragma unroll
    for (int t = 0; t < NT; ++t) {
        #pragma unroll
        for (int r = 0; r < 8; ++r) {
            unsigned int bits = __float_as_uint(bestd[t][r]);
            unsigned int mono = (bits & 0x80000000u) ? ~bits : (bits | 0x80000000u);
            unsigned long long v =
                ((unsigned long long)mono << 32) | (unsigned int)bestk[t][r];
            #pragma unroll
            for (int m = 1; m <= 8; m <<= 1) {
                unsigned long long o = shfl_xor_u64(v, m);
                v = (o < v) ? o : v;          // min dist, tie -> smallest k
            }
            if (laneN == 0) {
                int row = t * 16 + hi * 8 + r;
                cand[row * NWAVE + wave] = v;
            }
        }
    }
    __syncthreads();

    // ---- per-row argmin over the 16 wave winners; histogram ----
    if (tid < ROWS) {
        unsigned long long m = ~0ull;
        #pragma unroll
        for (int i = 0; i < NWAVE; ++i) {
            unsigned long long v = cand[tid * NWAVE + i];
            m = (v < m) ? v : m;
        }
        int kidx = (int)(m & 0xFFFFFFFFu);
        rowk[tid] = kidx;
        atomicAdd(&hist[kidx], 1.0f);
    }
    __syncthreads();

    // ---- gather codebook columns, write quantized(=quantized_st), accumulate SSE ----
    float lsum = 0.0f;
    for (int i = tid; i < ROWS * CDIM; i += BLK) {
        int r = i >> 9;
        int c = i & (CDIM - 1);
        float q  = emb[(size_t)c * KDIM + rowk[r]];
        float xv = xs[r * LDST + c];
        outq[(size_t)(m0 + r) * CDIM + c] = q;
        float dlt = q - xv;
        lsum += dlt * dlt;
    }
    red[tid] = lsum;
    __syncthreads();
    for (int s = BLK / 2; s > 0; s >>= 1) {
        if (tid < s) red[tid] += red[tid + s];
        __syncthreads();
    }
    if (tid == 0) atomicAdd(sse, red[0]);
}

// ---------------------------------------------------------------------------
// Kernel 3: finalize loss + perplexity
// ---------------------------------------------------------------------------
__global__ __launch_bounds__(256) void vq_finalize_kernel(
    const float* __restrict__ hist, const float* __restrict__ running,
    const float* __restrict__ sse, float* __restrict__ out) {
    __shared__ float red[256];
    const int tid = threadIdx.x;
    float s = 0.0f;
    for (int k = tid; k < KDIM; k += 256) {
        float avg = hist[k] * (1.0f / (float)MROWS);
        float p = 0.9f * running[k] + 0.1f * avg;   // EMA update (train=True)
        s += p * logf(p + 1e-10f);
    }
    red[tid] = s;
    __syncthreads();
    for (int off = 128; off > 0; off >>= 1) {
        if (tid < off) red[tid] += red[tid + off];
        __syncthreads();
    }
    if (tid == 0) {
        // q_latent_loss == e_latent_loss numerically -> loss = (1 + BETA) * mse
        out[(size_t)MROWS * CDIM]     = 1.25f * (*sse) / (float)((size_t)MROWS * CDIM);
        out[(size_t)MROWS * CDIM + 1] = expf(-red[0]);
    }
}

// ---------------------------------------------------------------------------
extern "C" void kernel_launch(void* const* d_in, const int* in_sizes, int n_in,
                              void* d_out, int out_size, void* d_ws, size_t ws_size,
                              hipStream_t stream) {
    const float* x       = (const float*)d_in[0];   // [B,N,C] = [8192, 512]
    const float* emb     = (const float*)d_in[1];   // [C,K]   = [512, 8192]
    const float* running = (const float*)d_in[2];   // [K]
    float* out = (float*)d_out;                     // quantized_st | loss | perplexity

    float* e2   = (float*)d_ws;      // [K]
    float* hist = e2 + KDIM;         // [K]
    float* sse  = hist + KDIM;       // [1]

    vq_zero_kernel<<<KDIM / 256, 256, 0, stream>>>(hist, sse);
    vq_e2_kernel<<<KDIM / 256, 256, 0, stream>>>(emb, e2);
    vq_main_kernel<<<MROWS / ROWS, BLK, 0, stream>>>(x, emb, e2, out, hist, sse);
    vq_finalize_kernel<<<1, 256, 0, stream>>>(hist, running, sse, out);
}